// ViSNeRF_5196910428444
// MI455X (gfx1250) — compile-verified
//
#include <hip/hip_runtime.h>

// ViSNeRF / K-planes multiresolution feature interpolation, MI455X (gfx1250).
// Gather-bound: all grids (~128MB) fit in the 192MB L2; output is 256MB and
// write-once -> stored non-temporally so it doesn't evict the L2-resident grids.
// wave32 == FEAT(32): one wave per point, lane == channel; coordinate math is
// wave-uniform, stores are one coalesced 128B line per level.
// Gathers use 32-bit element indices off uniform SGPR bases so codegen picks
// the saddr+voffset global_load form (minimal 64-bit VALU address math).

#define N_FEAT        32
#define N_LEVELS      4
#define T_RESO        150
#define PTS_PER_BLOCK 8

struct GridPtrs { const float* g[28]; };

#if __has_builtin(__builtin_amdgcn_global_load_async_to_lds_b32) && \
    __has_builtin(__builtin_amdgcn_s_wait_asynccnt)
#define HAVE_ASYNC_LDS 1
typedef __attribute__((address_space(1))) int as1_int;   // global b32
typedef __attribute__((address_space(3))) int as3_int;   // LDS b32
#else
#define HAVE_ASYNC_LDS 0
#endif

__device__ __forceinline__ float lerpf(float a, float b, float w) {
  return fmaf(w, b - a, a);
}

// r0/r1 = channel_offset + y*S (element index of row start), x0/x1 = column.
__device__ __forceinline__ float bilin(const float* __restrict__ b,
                                       unsigned r0, unsigned r1,
                                       unsigned x0, unsigned x1,
                                       float wx, float wy) {
  float v00 = b[r0 + x0];
  float v01 = b[r0 + x1];
  float v10 = b[r1 + x0];
  float v11 = b[r1 + x1];
  return lerpf(lerpf(v00, v01, wx), lerpf(v10, v11, wx), wy);
}

__global__ __launch_bounds__(256) void visnerf_feat_kernel(
    const float* __restrict__ pts, const float* __restrict__ ts,
    GridPtrs P, float* __restrict__ out, int n_pts)
{
  __shared__ float sp[32];   // 8 points x 3 coords (0..23) + 8 timestamps (24..31)
  const int  tid   = threadIdx.x;
  const int  lane  = tid & 31;
  const int  wave  = tid >> 5;
  const long baseN = (long)blockIdx.x * PTS_PER_BLOCK;

  if (tid < 32) {
    const float* src = (lane < 24) ? (pts + baseN * 3 + lane)
                                   : (ts + baseN + (lane - 24));
#if HAVE_ASYNC_LDS
    __builtin_amdgcn_global_load_async_to_lds_b32((as1_int*)src,
                                                  (as3_int*)&sp[lane], 0, 0);
#else
    sp[lane] = *src;
#endif
    if (tid == 0) {
      __builtin_prefetch(pts + (baseN + PTS_PER_BLOCK) * 3, 0, 1);
      __builtin_prefetch(ts + baseN + PTS_PER_BLOCK, 0, 1);
    }
  }
#if HAVE_ASYNC_LDS
  __builtin_amdgcn_s_wait_asynccnt(0);
#endif
  __syncthreads();

  const long n = baseN + wave;
  if (n >= n_pts) return;

  // p = -pts / BOUNDS ; p4 = [p0, p1, p2, t]
  const float p0 = -sp[wave * 3 + 0] / 1.3f;
  const float p1 = -sp[wave * 3 + 1] / 1.3f;
  const float p2 = -sp[wave * 3 + 2] / 1.3f;
  const float t  =  sp[24 + wave];

  // time axis interpolation (shared by all 4 levels)
  float ftf = fminf(fmaxf((t + 1.0f) * (0.5f * (float)(T_RESO - 1)), 0.0f),
                    (float)(T_RESO - 1));
  unsigned t0 = (unsigned)(int)ftf;
  unsigned t1 = (unsigned)min((int)t0 + 1, T_RESO - 1);
  float    wt = ftf - (float)t0;

  const unsigned c = (unsigned)lane;  // channel == lane (FEAT == warpSize == 32)
  const unsigned ct = c * T_RESO;
  float* __restrict__ outp = out + n * (long)(N_FEAT * N_LEVELS);

#pragma unroll
  for (int l = 0; l < N_LEVELS; ++l) {
    const int      S  = 64 << l;       // 64,128,256,512 (all 3 spatial axes equal)
    const unsigned uS = (unsigned)S;
    const float    hi = (float)(S - 1);
    const float    sm = 0.5f * hi;

    // per-axis [-1,1] -> [0,S-1] with border clamp (grid_sample align_corners=True)
    float f0 = fminf(fmaxf((p0 + 1.0f) * sm, 0.0f), hi);
    float f1 = fminf(fmaxf((p1 + 1.0f) * sm, 0.0f), hi);
    float f2 = fminf(fmaxf((p2 + 1.0f) * sm, 0.0f), hi);
    unsigned a0 = (unsigned)(int)f0, a1 = (unsigned)(int)f1, a2 = (unsigned)(int)f2;
    unsigned b0 = (unsigned)min((int)a0 + 1, S - 1);
    unsigned b1 = (unsigned)min((int)a1 + 1, S - 1);
    unsigned b2 = (unsigned)min((int)a2 + 1, S - 1);
    float w0 = f0 - (float)a0, w1 = f1 - (float)a1, w2 = f2 - (float)a2;

    const unsigned cpp = c * uS * uS;   // channel offset in a plane grid (elements)
    const unsigned cpl = c * uS;        // channel offset in a line grid  (elements)

    // Row-start element indices (channel offset folded in); columns added in bilin.
    const unsigned r0a1 = cpp + a1 * uS, r0b1 = cpp + b1 * uS;  // y from axis1
    const unsigned r0a2 = cpp + a2 * uS, r0b2 = cpp + b2 * uS;  // y from axis2

    // MAT = [(0,1),(0,2),(1,2)] : x <- p4[MAT[i][0]] (W dim), y <- p4[MAT[i][1]] (H dim)
    float plane0 = bilin(P.g[l * 7 + 0], r0a1, r0b1, a0, b0, w0, w1);
    float plane1 = bilin(P.g[l * 7 + 1], r0a2, r0b2, a0, b0, w0, w2);
    float plane2 = bilin(P.g[l * 7 + 2], r0a2, r0b2, a1, b1, w1, w2);

    // VEC = [2,1,0] : line i interpolates along axis VEC[i] (W==1 -> x=0, wx=0)
    const float* ln0 = P.g[l * 7 + 3];
    const float* ln1 = P.g[l * 7 + 4];
    const float* ln2 = P.g[l * 7 + 5];
    float line0 = lerpf(ln0[cpl + a2], ln0[cpl + b2], w2);
    float line1 = lerpf(ln1[cpl + a1], ln1[cpl + b1], w1);
    float line2 = lerpf(ln2[cpl + a0], ln2[cpl + b0], w0);

    const float* tmb = P.g[l * 7 + 6];
    float tv = lerpf(tmb[ct + t0], tmb[ct + t1], wt);

    float acc = plane0 * line0;
    acc = fmaf(plane1, line1, acc);
    acc = fmaf(plane2, line2, acc);

    // Write-once output stream: non-temporal so it doesn't evict L2-resident grids.
    __builtin_nontemporal_store(acc * tv, &outp[l * N_FEAT + (int)c]);
  }
}

extern "C" void kernel_launch(void* const* d_in, const int* in_sizes, int n_in,
                              void* d_out, int out_size, void* d_ws, size_t ws_size,
                              hipStream_t stream) {
  (void)d_ws; (void)ws_size; (void)out_size;
  const float* pts = (const float*)d_in[0];
  const float* ts  = (const float*)d_in[1];
  GridPtrs P;
  // d_in[2..29]: 4 levels x {plane01, plane02, plane12, line2, line1, line0, time}
  for (int i = 0; i < 28 && (2 + i) < n_in; ++i) P.g[i] = (const float*)d_in[2 + i];
  const int n_pts = in_sizes[1];  // timestamps count == N
  const int nblocks = (n_pts + PTS_PER_BLOCK - 1) / PTS_PER_BLOCK;
  visnerf_feat_kernel<<<nblocks, 256, 0, stream>>>(pts, ts, P, (float*)d_out, n_pts);
}